// QRLayer_65369402245694
// MI455X (gfx1250) — compile-verified
//
#include <hip/hip_runtime.h>

typedef float v2f __attribute__((ext_vector_type(2)));
typedef float v8f __attribute__((ext_vector_type(8)));

#define M_TOTAL 16384   // B*S = 4*4096
#define K_TOTAL 4096    // D_OUT
#define N_TOTAL 64      // D_IN == RANK
#define BK      64      // K tile staged in LDS
#define ROWS    64      // M rows per workgroup (4 waves * 16)
#define LDST    68      // padded LDS row stride in floats (272B, 16B aligned)

// ---------------------------------------------------------------------------
// Kernel 1: W = lora_B @ lora_A, fold SCALING (0.25) + rank mask, pack into
// the v_wmma_f32_16x16x4_f32 B-fragment lane layout:
//   chunk kb (K=4), tile t (N=16), lane = (K%4>=2)*16 + (N%16),
//   vgpr v = K&1  ->  wpack[((kb*4+t)*32 + lane)*2 + v]
// Also zeroes the absmax accumulator (re-initialized every call: graph-safe).
// ---------------------------------------------------------------------------
__global__ __launch_bounds__(256) void nf4_prep_kernel(
    const float* __restrict__ lora_A,   // [64,64]  (RANK x D_IN)
    const float* __restrict__ lora_B,   // [4096,64](D_OUT x RANK)
    const float* __restrict__ rank_imp, // [64]
    float* __restrict__ wpack,
    unsigned int* __restrict__ absmax_u)
{
    const int idx = blockIdx.x * blockDim.x + threadIdx.x;  // 0 .. 4096*64-1
    if (idx == 0) *absmax_u = 0u;
    const int k = idx >> 6;
    const int n = idx & 63;

    float acc = 0.f;
#pragma unroll 8
    for (int r = 0; r < 64; ++r)
        acc = fmaf(lora_B[k * 64 + r], lora_A[r * 64 + n], acc);

    // sigmoid(ri) > 0.1  <=>  ri > logit(0.1)
    const float THR = -2.1972245773362196f;
    bool any = false;
    for (int j = 0; j < 64; ++j) any = any || (rank_imp[j] > THR);
    const float m = any ? ((rank_imp[n] > THR) ? 1.f : 0.f) : 1.f;
    const float w = acc * 0.25f * m;   // SCALING = ALPHA/RANK = 0.25

    const int kb = k >> 2, ko = k & 3;
    const int vsel = ko & 1, lh = ko >> 1;
    const int t = n >> 4, nl = n & 15;
    const int lanep = lh * 16 + nl;
    wpack[(((size_t)kb * 4 + t) * 32 + lanep) * 2 + vsel] = w;
}

// ---------------------------------------------------------------------------
// Kernel 2: Y = x @ Wmasked  via V_WMMA_F32_16X16X4_F32.
// 4 waves per block; each wave computes a 16x64 tile (4 v8f accumulators).
// x tiles double-buffered: next tile prefetched into registers while the
// current LDS tile feeds the WMMAs, keeping HBM loads in flight (the kernel
// is bandwidth-bound: 268 MB of x at 23.3 TB/s ~ 11.7 us floor).
// Epilogue: store Y + deterministic global absmax (uint-ordered atomicMax).
// ---------------------------------------------------------------------------
__global__ __launch_bounds__(128) void nf4_gemm_kernel(
    const float* __restrict__ x,       // [16384, 4096]
    const float* __restrict__ wpack,   // packed B fragments
    float* __restrict__ y,             // [16384, 64]
    unsigned int* __restrict__ absmax_u)
{
    __shared__ float lds[ROWS * LDST];
    __shared__ float red[128];

    const int tid  = threadIdx.x;
    const int wave = tid >> 5;
    const int lane = tid & 31;
    const int m0   = blockIdx.x * ROWS;

    v8f acc0 = {}, acc1 = {}, acc2 = {}, acc3 = {};

    // cooperative x-tile load mapping: 64 rows x 64 cols, 8 x float4 / thread
    const int lr = tid >> 4;           // 0..7  (row group)
    const int lc = (tid & 15) << 2;    // 0,4,...,60 (col, floats)
    const float* xbase = x + (size_t)(m0 + lr) * K_TOTAL + lc;
    float*       sbase = &lds[lr * LDST + lc];

    // A-fragment read base: within this wave's 16-row slice,
    // lane l reads row (l&15), K offset 2*(l>>4) + 4c
    const int arow_off = wave * 16 * LDST + (lane & 15) * LDST + 2 * (lane >> 4);

    // ---- prefetch first x tile into registers ----
    float4 stage[8];
#pragma unroll
    for (int s = 0; s < 8; ++s)
        stage[s] = *(const float4*)(xbase + (size_t)(s << 3) * K_TOTAL);

    for (int k0 = 0; k0 < K_TOTAL; k0 += BK) {
        __syncthreads();               // previous compute done reading LDS
#pragma unroll
        for (int s = 0; s < 8; ++s)
            *(float4*)(sbase + (s << 3) * LDST) = stage[s];
        __syncthreads();               // tile visible to all waves

        // issue next tile's global loads now; they stay outstanding while the
        // WMMAs below execute (waited at the ds_store of the next iteration)
        if (k0 + BK < K_TOTAL) {
            const float* xn = xbase + k0 + BK;
#pragma unroll
            for (int s = 0; s < 8; ++s)
                stage[s] = *(const float4*)(xn + (size_t)(s << 3) * K_TOTAL);
        }

        const float* bbase = wpack + (size_t)(k0 >> 2) * 256 + lane * 2;
#pragma unroll
        for (int c = 0; c < 16; ++c) {   // 16 K-chunks of 4
            const v2f a = *(const v2f*)(&lds[arow_off + (c << 2)]);
            const float* bp = bbase + (size_t)c * 256;
            const v2f b0 = *(const v2f*)(bp);
            const v2f b1 = *(const v2f*)(bp + 64);
            const v2f b2 = *(const v2f*)(bp + 128);
            const v2f b3 = *(const v2f*)(bp + 192);
            acc0 = __builtin_amdgcn_wmma_f32_16x16x4_f32(false, a, false, b0,
                                                         (short)0, acc0, false, false);
            acc1 = __builtin_amdgcn_wmma_f32_16x16x4_f32(false, a, false, b1,
                                                         (short)0, acc1, false, false);
            acc2 = __builtin_amdgcn_wmma_f32_16x16x4_f32(false, a, false, b2,
                                                         (short)0, acc2, false, false);
            acc3 = __builtin_amdgcn_wmma_f32_16x16x4_f32(false, a, false, b3,
                                                         (short)0, acc3, false, false);
        }
    }

    // C/D layout: VGPR v, lanes 0-15 -> M=v, N=lane; lanes 16-31 -> M=v+8
    const int mrow = m0 + wave * 16 + 8 * (lane >> 4);
    const int ncol = lane & 15;
    float lmax = 0.f;
#pragma unroll
    for (int v = 0; v < 8; ++v) {
        const float e0 = acc0[v], e1 = acc1[v], e2 = acc2[v], e3 = acc3[v];
        float* yp = y + (size_t)(mrow + v) * N_TOTAL + ncol;
        yp[0] = e0; yp[16] = e1; yp[32] = e2; yp[48] = e3;
        lmax = fmaxf(lmax, fmaxf(fmaxf(fabsf(e0), fabsf(e1)),
                                 fmaxf(fabsf(e2), fabsf(e3))));
    }

    red[tid] = lmax;
    __syncthreads();
    for (int off = 64; off > 0; off >>= 1) {
        if (tid < off) red[tid] = fmaxf(red[tid], red[tid + off]);
        __syncthreads();
    }
    if (tid == 0) atomicMax(absmax_u, __float_as_uint(red[0]));
}

// ---------------------------------------------------------------------------
// Kernel 3: NF4 snap with global absmax, * quantization_scale. In-place on y.
// ---------------------------------------------------------------------------
__global__ __launch_bounds__(256) void nf4_quant_kernel(
    float* __restrict__ y,
    const float* __restrict__ qscale,
    const unsigned int* __restrict__ absmax_u,
    int n4)
{
    const int i = blockIdx.x * blockDim.x + threadIdx.x;
    if (i >= n4) return;

    const float am   = __uint_as_float(*absmax_u);
    const float safe = (am == 0.f) ? 1.f : am;
    const float qs   = qscale[0];

    const float LV[16] = {
        -1.0f, -0.6961928009986877f, -0.5250730514526367f, -0.39491748809814453f,
        -0.28444138169288635f, -0.18477343022823334f, -0.09105003625154495f, 0.0f,
        0.07958029955625534f, 0.16093020141124725f, 0.24611230194568634f,
        0.33791524171829224f, 0.44070982933044434f, 0.5626170039176941f,
        0.7229568362236023f, 1.0f};

    float4 v = ((const float4*)y)[i];
    float in[4] = {v.x, v.y, v.z, v.w};
    float out[4];
#pragma unroll
    for (int e = 0; e < 4; ++e) {
        const float nrm = in[e] / safe;
        float best = LV[0];
        float bd   = fabsf(nrm - LV[0]);
#pragma unroll
        for (int j = 1; j < 16; ++j) {
            const float d = fabsf(nrm - LV[j]);
            if (d < bd) { bd = d; best = LV[j]; }   // first-min tie-break == argmin
        }
        const float q = best * am;
        out[e] = ((am == 0.f) ? in[e] : q) * qs;
    }
    float4 r; r.x = out[0]; r.y = out[1]; r.z = out[2]; r.w = out[3];
    ((float4*)y)[i] = r;
}

// ---------------------------------------------------------------------------
extern "C" void kernel_launch(void* const* d_in, const int* in_sizes, int n_in,
                              void* d_out, int out_size, void* d_ws, size_t ws_size,
                              hipStream_t stream)
{
    const float* x      = (const float*)d_in[0];
    const float* lora_A = (const float*)d_in[1];
    const float* lora_B = (const float*)d_in[2];
    const float* qscale = (const float*)d_in[3];
    const float* rimp   = (const float*)d_in[4];
    float* out = (float*)d_out;

    unsigned int* absmax_u = (unsigned int*)d_ws;
    float* wpack = (float*)((char*)d_ws + 256);   // 4096*64 floats = 1 MB

    nf4_prep_kernel<<<(K_TOTAL * N_TOTAL) / 256, 256, 0, stream>>>(
        lora_A, lora_B, rimp, wpack, absmax_u);

    nf4_gemm_kernel<<<M_TOTAL / ROWS, 128, 0, stream>>>(
        x, wpack, out, absmax_u);

    const int n4 = out_size / 4;   // 262144 float4s
    nf4_quant_kernel<<<(n4 + 255) / 256, 256, 0, stream>>>(
        out, qscale, absmax_u, n4);
}